// HeatSolver_67929202753588
// MI455X (gfx1250) — compile-verified
//
#include <hip/hip_runtime.h>
#include <cstdint>

// Problem constants (match the JAX reference)
#define GNX    512
#define GSTEPS 512
#define GNTRAJ 16384

// f32-rounded constants exactly as the reference produces them
#define DT_F32  ((float)(0.01 / 511.0))                    // delta_t
#define DX2_F32 ((float)((6.0 / 511.0) * (6.0 / 511.0)))   // delta_x**2
#define DXG_F32 ((float)(6.0 / 511.0))                     // grid spacing

// Temporal-blocking geometry
#define NSUB_MAX 8                  // sub-steps per mega launch
#define RSZ 32                      // staged region: 16 core + 2*8 halo
#define RST 34                      // padded LDS row stride (banks)

// ---------------------------------------------------------------------------
// CDNA5 async global->LDS staging (ASYNCcnt path, gfx1250-specific)
// ---------------------------------------------------------------------------
__device__ __forceinline__ void async_load_lds_b32(unsigned lds_byte_off,
                                                   const float* gsrc) {
  asm volatile("global_load_async_to_lds_b32 %0, %1, off"
               :: "v"(lds_byte_off),
                  "v"((unsigned long long)(uintptr_t)gsrc)
               : "memory");
}

__device__ __forceinline__ void wait_async_zero() {
#if __has_builtin(__builtin_amdgcn_s_wait_asynccnt)
  __builtin_amdgcn_s_wait_asynccnt(0);
#else
  asm volatile("s_wait_asynccnt 0" ::: "memory");
#endif
}

// ---------------------------------------------------------------------------
// Bilinear interpolation of one trajectory point (mirrors reference math)
// ---------------------------------------------------------------------------
__device__ __forceinline__ float interp_point(const float* __restrict__ u,
                                              const float* __restrict__ tgt3) {
  float x = tgt3[0];
  float y = tgt3[1];
  float xi = fminf(fmaxf((x + 3.0f) / DXG_F32, 0.0f), 511.0f);
  float yi = fminf(fmaxf((y + 3.0f) / DXG_F32, 0.0f), 511.0f);
  int x0 = (int)floorf(xi);
  int y0 = (int)floorf(yi);
  int x1 = min(x0 + 1, GNX - 1);
  int y1 = min(y0 + 1, GNX - 1);
  float Q11 = u[y0 * GNX + x0];
  float Q21 = u[y0 * GNX + x1];
  float Q12 = u[y1 * GNX + x0];
  float Q22 = u[y1 * GNX + x1];
  float x0f = (float)x0, x1f = (float)x1;
  float y0f = (float)y0, y1f = (float)y1;
  return Q11 * (x1f - xi) * (y1f - yi)
       + Q21 * (xi - x0f) * (y1f - yi)
       + Q12 * (x1f - xi) * (yi - y0f)
       + Q22 * (xi - x0f) * (yi - y0f);
}

// ---------------------------------------------------------------------------
// Kernel 1: slice0 = u0 with boundary ring forced to zero
// ---------------------------------------------------------------------------
__global__ __launch_bounds__(256) void heat_init(const float* __restrict__ u0,
                                                 float* __restrict__ s0) {
  int idx = blockIdx.x * 256 + threadIdx.x;   // 0 .. 512*512-1
  int y = idx >> 9;
  int x = idx & (GNX - 1);
  float v = u0[idx];
  if (x == 0 || x == GNX - 1 || y == 0 || y == GNX - 1) v = 0.0f;
  s0[idx] = v;
}

// ---------------------------------------------------------------------------
// Kernel 2: temporally-blocked FTCS mega-step.
// Each block: async-stage a 32x32 (clamped) region of `src` into LDS, run
// `nsub` (<= 8) sub-steps in LDS, and after each sub-step store its 16x16
// core of slice t+s into bank[s-1]. All 1024 cores tile the grid exactly.
// ---------------------------------------------------------------------------
__global__ __launch_bounds__(256) void heat_mega(
    const float* __restrict__ src,      // full slice t
    float* __restrict__ bank,           // nsub full slices: t+1 .. t+nsub
    const float* __restrict__ alpha_p,
    int nsub)
{
  __shared__ float buf0[RSZ * RST];
  __shared__ float buf1[RSZ * RST];

  const int tid = threadIdx.x;                   // 0..255
  const int bx = blockIdx.x, by = blockIdx.y;    // 32 x 32 blocks
  const int oy = by * 16 - 8;                    // region origin (global)
  const int ox = bx * 16 - 8;

  // --- async stage 32x32 region (addresses clamped in-grid) ---------------
  #pragma unroll
  for (int k = tid; k < RSZ * RSZ; k += 256) {
    int r = k >> 5, c = k & 31;
    int sy = min(max(oy + r, 0), GNX - 1);
    int sx = min(max(ox + c, 0), GNX - 1);
    async_load_lds_b32((unsigned)(uintptr_t)(&buf0[r * RST + c]),
                       src + sy * GNX + sx);
  }
  wait_async_zero();          // each wave drains its own ASYNCcnt
  __syncthreads();            // all waves' LDS writes visible

  const float gamma = alpha_p[0] * DT_F32 / DX2_F32;

  float* cur = buf0;
  float* nxt = buf1;
  // core cell owned by this thread (one per thread)
  const int cr = 8 + (tid >> 4);
  const int cc = 8 + (tid & 15);
  const int cgy = oy + cr, cgx = ox + cc;

  for (int s = 1; s <= nsub; ++s) {
    // compute interior of region; physical-boundary rule per cell.
    for (int k = tid; k < 30 * 30; k += 256) {
      int r = 1 + k / 30;
      int c = 1 + k % 30;
      int gy = oy + r, gx = ox + c;
      float v = 0.0f;
      if (gy >= 1 && gy <= GNX - 2 && gx >= 1 && gx <= GNX - 2) {
        float cen = cur[r * RST + c];
        float dn  = cur[(r + 1) * RST + c];   // u[i+1, j]
        float up  = cur[(r - 1) * RST + c];   // u[i-1, j]
        float rt  = cur[r * RST + c + 1];     // u[i, j+1]
        float lf  = cur[r * RST + c - 1];     // u[i, j-1]
        float lap = (((dn + up) + rt) + lf) - 4.0f * cen;
        v = cen + gamma * lap;
      }
      nxt[r * RST + c] = v;
    }
    __syncthreads();   // nxt fully written before core store

    // store this block's 16x16 core of slice t+s
    bank[(size_t)(s - 1) * GNX * GNX + cgy * GNX + cgx] = nxt[cr * RST + cc];

    __syncthreads();   // defensive: stores done before cur gets overwritten

    float* tmp = cur; cur = nxt; nxt = tmp;
  }
}

// ---------------------------------------------------------------------------
// Kernel 3: interpolate nsub consecutive slices (64 blocks per slice)
// ---------------------------------------------------------------------------
__global__ __launch_bounds__(256) void interp_multi(
    const float* __restrict__ bank,     // nsub full slices
    const float* __restrict__ tgt_base, // target + t0*GNTRAJ*3
    float* __restrict__ out_base)       // out    + t0*GNTRAJ
{
  int s = blockIdx.x >> 6;                       // slice within bank
  int n = (blockIdx.x & 63) * 256 + threadIdx.x; // point 0..16383
  const float* u = bank + (size_t)s * GNX * GNX;
  size_t idx = (size_t)s * GNTRAJ + n;
  out_base[idx] = interp_point(u, tgt_base + idx * 3);
}

// ---------------------------------------------------------------------------
// Host entry
// ---------------------------------------------------------------------------
extern "C" void kernel_launch(void* const* d_in, const int* in_sizes, int n_in,
                              void* d_out, int out_size, void* d_ws, size_t ws_size,
                              hipStream_t stream) {
  (void)in_sizes; (void)n_in; (void)out_size;

  const float* u0     = (const float*)d_in[0];   // (512, 512) f32
  const float* target = (const float*)d_in[1];   // (512, 16384, 3) f32
  const float* alpha  = (const float*)d_in[2];   // scalar f32
  float* out = (float*)d_out;                    // (512, 16384) f32

  const size_t slice = (size_t)GNX * GNX;        // elements per slice

  // Workspace: slice0 + 2 banks of ns_max slices. Degrade ns_max if ws small.
  size_t ws_slices = ws_size / (slice * sizeof(float));
  int ns_max;
  if (ws_slices >= 1 + 2 * NSUB_MAX)      ns_max = NSUB_MAX;
  else if (ws_slices >= 3)                ns_max = (int)((ws_slices - 1) / 2);
  else                                    ns_max = 1;   // minimal assumption
  if (ns_max > NSUB_MAX) ns_max = NSUB_MAX;

  float* s0    = (float*)d_ws;
  float* bankA = s0 + slice;
  float* bankB = bankA + (size_t)ns_max * slice;

  // slice 0 = boundary-zeroed u0, then its interpolation (out row 0)
  heat_init<<<dim3((GNX * GNX) / 256), dim3(256), 0, stream>>>(u0, s0);
  interp_multi<<<dim3(64), dim3(256), 0, stream>>>(s0, target, out);

  // mega-steps: each advances ns <= ns_max slices and interpolates them
  const float* prev = s0;
  int b = 0;
  int t = 0;                                     // last completed slice index
  while (t < GSTEPS - 1) {
    int ns = GSTEPS - 1 - t;
    if (ns > ns_max) ns = ns_max;
    float* bk = b ? bankB : bankA;

    heat_mega<<<dim3(GNX / 16, GNX / 16), dim3(256), 0, stream>>>(
        prev, bk, alpha, ns);
    interp_multi<<<dim3(ns * 64), dim3(256), 0, stream>>>(
        bk,
        target + (size_t)(t + 1) * GNTRAJ * 3,
        out + (size_t)(t + 1) * GNTRAJ);

    prev = bk + (size_t)(ns - 1) * slice;
    b ^= 1;
    t += ns;
  }
}